// LipSeqLoss_609885356924
// MI455X (gfx1250) — compile-verified
//
#include <hip/hip_runtime.h>
#include <hip/hip_bf16.h>

// Problem constants from the reference: B=256, T=64, C=2000.
#define LSL_B 256
#define LSL_T 64
#define LSL_C 2000

typedef __attribute__((ext_vector_type(2))) float v2f;
typedef __attribute__((ext_vector_type(8))) float v8f;

// One block of 256 threads (8 wave32 waves).
// Phase 1: each thread gathers -log p[target] at the last valid timestep into LDS.
// Phase 2: wave 0 reduces the 256 values exactly with V_WMMA_F32_16X16X4_F32
//          (A = ones, so D accumulates column sums of B; 4 chunks of 64 values),
//          then a 5-step wave32 xor-shuffle collapses the 16 column sums.
__global__ void __launch_bounds__(LSL_B)
LipSeqLoss_kernel(const float* __restrict__ inp,
                  const int* __restrict__ length,
                  const int* __restrict__ target,
                  float* __restrict__ out) {
    __shared__ float vals[LSL_B];

    const int b = threadIdx.x;          // 0..255, one batch row per thread
    int len = length[b];
    if (len > LSL_T) len = LSL_T;
    const int t   = len - 1;            // last valid timestep
    const int tgt = target[b];

    // gather: input[b, t, tgt]
    const size_t idx = ((size_t)b * LSL_T + (size_t)t) * LSL_C + (size_t)tgt;
    vals[b] = inp[idx];

    __syncthreads();

    // Uniform branch: wave 0 has all 32 lanes active => EXEC all ones for WMMA.
    if ((threadIdx.x >> 5) == 0) {
        const int lane = threadIdx.x;   // 0..31

        // A = 16x4 ones matrix (2 VGPRs per lane).
        v2f a; a.x = 1.0f; a.y = 1.0f;
        v8f c = {};                      // accumulator starts at zero

        // Four accumulating WMMAs, each consuming 64 LDS values.
        // B is 4x16 f32 = exactly 2 VGPRs x 32 lanes = 64 slots; any bijective
        // packing of 64 values into B yields the same total column-sum mass.
        #pragma unroll
        for (int chunk = 0; chunk < 4; ++chunk) {
            v2f bm;
            bm.x = vals[chunk * 64 + lane];
            bm.y = vals[chunk * 64 + 32 + lane];
            // 8 args: (neg_a, A, neg_b, B, c_mod, C, reuse_a, reuse_b)
            c = __builtin_amdgcn_wmma_f32_16x16x4_f32(
                    /*neg_a=*/false, a,
                    /*neg_b=*/false, bm,
                    /*c_mod=*/(short)0, c,
                    /*reuse_a=*/false, /*reuse_b=*/false);
        }

        // D row m is identical for all m: D[m][n] = colsum[n].
        // c[0] (VGPR0): lanes 0-15 hold colsum[0..15] (M=0), lanes 16-31 hold
        // colsum[0..15] again (M=8) -> sum over 32 lanes == 2 * total.
        float s = c[0];
        #pragma unroll
        for (int off = 16; off > 0; off >>= 1)
            s += __shfl_xor(s, off, 32);

        if (lane == 0) {
            // nll = -gathered; mask.sum() == B == 256; doubled by lane replication.
            out[0] = -s * (0.5f / (float)LSL_B);
        }
    }
}

extern "C" void kernel_launch(void* const* d_in, const int* in_sizes, int n_in,
                              void* d_out, int out_size, void* d_ws, size_t ws_size,
                              hipStream_t stream) {
    (void)in_sizes; (void)n_in; (void)d_ws; (void)ws_size; (void)out_size;
    const float* inp    = (const float*)d_in[0];
    const int*   length = (const int*)d_in[1];
    const int*   target = (const int*)d_in[2];
    float*       out    = (float*)d_out;

    LipSeqLoss_kernel<<<1, LSL_B, 0, stream>>>(inp, length, target, out);
}